// AAMSoftmax_39676907888373
// MI455X (gfx1250) — compile-verified
//
#include <hip/hip_runtime.h>
#include <hip/hip_bf16.h>

// ---------------------------------------------------------------------------
// AAM-Softmax (ArcFace margin) for MI455X / gfx1250, wave32, f32 WMMA.
//   B=1024 rows, D=256 features, C=100000 classes.
//   out = 50 * (normalize(x) @ normalize(W)^T), with margin on target class.
// Roofline: output write 409.6 MB + weight read 102.4 MB (fits in 192 MB L2)
// => ~22 us HBM floor; f32 WMMA keeps full reference precision.
// ---------------------------------------------------------------------------

typedef __attribute__((ext_vector_type(2))) float v2f;
typedef __attribute__((ext_vector_type(8))) float v8f;

#define BB 1024
#define DD 256
#define CC 100000

#define SCALE_S 50.0f
#define COS_M 0.8775825618903728f
#define SIN_M 0.4794255386042030f
#define NEPS 1e-12f

#define WMMA_F32(A, Bv, Cv) \
    __builtin_amdgcn_wmma_f32_16x16x4_f32(false, (A), false, (Bv), (short)0, (Cv), false, false)

// ---------------------------------------------------------------------------
// Kernel 1: L2-normalize input rows.  One wave per row; 8 waves per block.
// ---------------------------------------------------------------------------
__global__ void norm_input_kernel(const float* __restrict__ in,
                                  float* __restrict__ nin) {
    const int lane = threadIdx.x & 31;
    const int wave = threadIdx.x >> 5;
    const int row  = blockIdx.x * 8 + wave;

    const float* rp = in  + (size_t)row * DD + lane * 8;
    float*       wp = nin + (size_t)row * DD + lane * 8;

    float v[8];
    float s = 0.f;
#pragma unroll
    for (int i = 0; i < 8; ++i) { v[i] = rp[i]; s += v[i] * v[i]; }

#pragma unroll
    for (int m = 16; m >= 1; m >>= 1) s += __shfl_xor(s, m, 32);

    const float inv = 1.0f / fmaxf(sqrtf(s), NEPS);
#pragma unroll
    for (int i = 0; i < 8; ++i) wp[i] = v[i] * inv;
}

// ---------------------------------------------------------------------------
// Kernel 2: per-row inverse L2 norm of weight.  One wave per row.
// ---------------------------------------------------------------------------
__global__ void weight_invnorm_kernel(const float* __restrict__ w,
                                      float* __restrict__ invw) {
    const int lane = threadIdx.x & 31;
    const int wave = threadIdx.x >> 5;
    const int row  = blockIdx.x * 8 + wave;

    const float* rp = w + (size_t)row * DD + lane * 8;
    float s = 0.f;
#pragma unroll
    for (int i = 0; i < 8; ++i) { const float t = rp[i]; s += t * t; }
#pragma unroll
    for (int m = 16; m >= 1; m >>= 1) s += __shfl_xor(s, m, 32);

    if (lane == 0) invw[row] = 1.0f / fmaxf(sqrtf(s), NEPS);
}

// ---------------------------------------------------------------------------
// Kernel 3: f32 WMMA GEMM.
//   out[b][c] = S * invW[c] * sum_k nIn[b][k] * W[c][k]
// Block = 128 threads = 4 waves.  Wave tile = 64 rows x 32 cols
// (4x2 WMMA tiles: 8 v_wmma per 6 fragment loads, 48 B cache traffic per
// output element).  Block tile = 256 rows x 32 cols.  Grid = (C/32, B/256).
//
// A fragment (16x4 f32, 2 VGPRs): lanes 0-15 hold K={k,k+1}, lanes 16-31
// K={k+2,k+3} of row M=lane%16 -> v2f load at (row, k + 2*(lane>>4)).
// B fragment mirrors A with lanes indexing N (weight rows are B columns).
// C/D (16x16 f32, 8 VGPRs): reg r, lane L -> M = r + 8*(L>>4), N = L%16.
// ---------------------------------------------------------------------------
__global__ void aam_gemm_kernel(const float* __restrict__ nin,
                                const float* __restrict__ w,
                                const float* __restrict__ invw,
                                float* __restrict__ out) {
    const int lane = threadIdx.x & 31;
    const int wave = threadIdx.x >> 5;
    const int half = lane >> 4;   // 0: K lo pair, 1: K hi pair
    const int l    = lane & 15;   // M index (A) / N index (B)

    const int rowBase = blockIdx.y * 256 + wave * 64;
    const int colBase = blockIdx.x * 32;

    const float* a0 = nin + (size_t)(rowBase + l) * DD + 2 * half;
    const float* a1 = a0 + (size_t)16 * DD;
    const float* a2 = a0 + (size_t)32 * DD;
    const float* a3 = a0 + (size_t)48 * DD;
    const float* b0 = w   + (size_t)(colBase + l) * DD + 2 * half;
    const float* b1 = b0 + (size_t)16 * DD;

    v8f acc00 = {}, acc01 = {};
    v8f acc10 = {}, acc11 = {};
    v8f acc20 = {}, acc21 = {};
    v8f acc30 = {}, acc31 = {};

#pragma unroll 4
    for (int k = 0; k < DD; k += 4) {
        const v2f A0 = *(const v2f*)(a0 + k);
        const v2f A1 = *(const v2f*)(a1 + k);
        const v2f A2 = *(const v2f*)(a2 + k);
        const v2f A3 = *(const v2f*)(a3 + k);
        const v2f B0 = *(const v2f*)(b0 + k);
        const v2f B1 = *(const v2f*)(b1 + k);
        acc00 = WMMA_F32(A0, B0, acc00);
        acc01 = WMMA_F32(A0, B1, acc01);
        acc10 = WMMA_F32(A1, B0, acc10);
        acc11 = WMMA_F32(A1, B1, acc11);
        acc20 = WMMA_F32(A2, B0, acc20);
        acc21 = WMMA_F32(A2, B1, acc21);
        acc30 = WMMA_F32(A3, B0, acc30);
        acc31 = WMMA_F32(A3, B1, acc31);
    }

    // Epilogue: fold in S * invW[col]; scatter D tiles (coalesced across l).
    const int col0 = colBase + l;
    const int col1 = colBase + 16 + l;
    const float s0 = SCALE_S * invw[col0];
    const float s1 = SCALE_S * invw[col1];

    const int r0 = rowBase +      half * 8;
    const int r1 = rowBase + 16 + half * 8;
    const int r2 = rowBase + 32 + half * 8;
    const int r3 = rowBase + 48 + half * 8;

#pragma unroll
    for (int r = 0; r < 8; ++r) {
        out[(size_t)(r0 + r) * CC + col0] = acc00[r] * s0;
        out[(size_t)(r0 + r) * CC + col1] = acc01[r] * s1;
        out[(size_t)(r1 + r) * CC + col0] = acc10[r] * s0;
        out[(size_t)(r1 + r) * CC + col1] = acc11[r] * s1;
        out[(size_t)(r2 + r) * CC + col0] = acc20[r] * s0;
        out[(size_t)(r2 + r) * CC + col1] = acc21[r] * s1;
        out[(size_t)(r3 + r) * CC + col0] = acc30[r] * s0;
        out[(size_t)(r3 + r) * CC + col1] = acc31[r] * s1;
    }
}

// ---------------------------------------------------------------------------
// Kernel 4: margin fixup on the target-class entry of each row.
// ---------------------------------------------------------------------------
__global__ void margin_fixup_kernel(const int* __restrict__ label,
                                    float* __restrict__ out) {
    const int b = blockIdx.x * blockDim.x + threadIdx.x;
    if (b >= BB) return;
    const int c = label[b];
    const size_t idx = (size_t)b * CC + c;
    const float t = out[idx] * (1.0f / SCALE_S);     // recover cosine
    const float sine = sqrtf(fmaxf(1.0f - t * t, 0.0f));
    const float phi  = t * COS_M - sine * SIN_M;
    const float nt   = (t > 0.0f) ? phi : t;
    out[idx] = SCALE_S * nt;
}

// ---------------------------------------------------------------------------
extern "C" void kernel_launch(void* const* d_in, const int* in_sizes, int n_in,
                              void* d_out, int out_size, void* d_ws, size_t ws_size,
                              hipStream_t stream) {
    const float* d_input  = (const float*)d_in[0];  // [B, D]
    const int*   d_label  = (const int*)d_in[1];    // [B]
    const float* d_weight = (const float*)d_in[2];  // [C, D]
    float*       out      = (float*)d_out;          // [B, C]

    // Workspace layout: normalized input [B*D] f32, then invW [C] f32.
    float* nin  = (float*)d_ws;
    float* invw = nin + (size_t)BB * DD;

    // 1) normalize input rows: 8 rows/block, 128 blocks
    norm_input_kernel<<<dim3(BB / 8), dim3(256), 0, stream>>>(d_input, nin);

    // 2) weight inverse norms: 8 rows/block, 12500 blocks
    weight_invnorm_kernel<<<dim3(CC / 8), dim3(256), 0, stream>>>(d_weight, invw);

    // 3) WMMA GEMM: grid (C/32, B/256) = (3125, 4), 128 threads (4 waves)
    aam_gemm_kernel<<<dim3(CC / 32, BB / 256), dim3(128), 0, stream>>>(
        nin, d_weight, invw, out);

    // 4) margin fixup: 1024 target entries
    margin_fixup_kernel<<<dim3(4), dim3(256), 0, stream>>>(d_label, out);
}